// ECELoss_2207613190488
// MI455X (gfx1250) — compile-verified
//
#include <hip/hip_runtime.h>

#define N_BINS 15
#define HIST_BINS 16          // padded to 16 for WMMA-friendly layout
#define HIST_COPIES 32        // one copy per lane -> conflict-free ds atomics in a wave
#define GRID1 2048
#define BLOCK1 256

typedef __attribute__((ext_vector_type(2))) float v2f;
typedef __attribute__((ext_vector_type(8))) float v8f;

// ---------------------------------------------------------------------------
// Stage 1: streaming histogram. Memory-bound: 256 MB @ 23.3 TB/s ~= 11 us.
// Coalesced b128 loads, sigmoid in VALU, 2 LDS atomics per element
// (ds_add_f32 for sum_p, ds_add_u64 packing {count:hi32, sum_t:lo32}).
// Emits per-workgroup partials: [GRID1][48] floats = {sum_p[16],count[16],sum_t[16]}.
// ---------------------------------------------------------------------------
__global__ __launch_bounds__(BLOCK1) void ece_partials_kernel(
    const float* __restrict__ logits, const int* __restrict__ targets,
    float* __restrict__ partials, int n)
{
    __shared__ float              s_p[HIST_COPIES][HIST_BINS];
    __shared__ unsigned long long s_ct[HIST_COPIES][HIST_BINS];

    const int tid = threadIdx.x;
    for (int i = tid; i < HIST_COPIES * HIST_BINS; i += BLOCK1) {
        (&s_p[0][0])[i]  = 0.0f;
        (&s_ct[0][0])[i] = 0ULL;
    }
    __syncthreads();

    const int lane = tid & 31;
    const int n4 = n >> 2;
    const long long stride = (long long)GRID1 * BLOCK1;
    const float4* lg4 = (const float4*)logits;
    const int4*   tg4 = (const int4*)targets;

    for (long long i = (long long)blockIdx.x * BLOCK1 + tid; i < n4; i += stride) {
        if (i + stride < n4) {
            __builtin_prefetch(&lg4[i + stride], 0, 1);   // global_prefetch_b8
            __builtin_prefetch(&tg4[i + stride], 0, 1);
        }
        float4 lg = lg4[i];   // global_load_b128
        int4   tg = tg4[i];   // global_load_b128
        #pragma unroll
        for (int j = 0; j < 4; ++j) {
            float x = (j == 0) ? lg.x : (j == 1) ? lg.y : (j == 2) ? lg.z : lg.w;
            int   t = (j == 0) ? tg.x : (j == 1) ? tg.y : (j == 2) ? tg.z : tg.w;
            float p = 1.0f / (1.0f + __expf(-x));
            int bin = (int)ceilf(p * (float)N_BINS) - 1;
            bin = bin < 0 ? 0 : (bin > N_BINS - 1 ? N_BINS - 1 : bin);
            atomicAdd(&s_p[lane][bin], p);                                   // ds_add_f32
            atomicAdd(&s_ct[lane][bin],
                      (1ULL << 32) | (unsigned long long)(unsigned)t);       // ds_add_u64
        }
    }

    // scalar tail (n not multiple of 4); no-op for the 2^25 reference size
    if (blockIdx.x == 0 && tid == 0) {
        for (int i = n4 * 4; i < n; ++i) {
            float p = 1.0f / (1.0f + __expf(-logits[i]));
            int bin = (int)ceilf(p * (float)N_BINS) - 1;
            bin = bin < 0 ? 0 : (bin > N_BINS - 1 ? N_BINS - 1 : bin);
            atomicAdd(&s_p[0][bin], p);
            atomicAdd(&s_ct[0][bin],
                      (1ULL << 32) | (unsigned long long)(unsigned)targets[i]);
        }
    }
    __syncthreads();

    if (tid < HIST_BINS) {
        float sp = 0.0f;
        unsigned long long ct = 0ULL;
        for (int c = 0; c < HIST_COPIES; ++c) { sp += s_p[c][tid]; ct += s_ct[c][tid]; }
        float cnt = (float)(unsigned)(ct >> 32);
        float st  = (float)(unsigned)(ct & 0xffffffffULL);
        float* out = partials + (size_t)blockIdx.x * 48;
        out[tid]      = sp;
        out[16 + tid] = cnt;
        out[32 + tid] = st;
    }
}

// ---------------------------------------------------------------------------
// Stage 2: one wave32 reduces [GRID1][48] partials with V_WMMA_F32_16X16X4_F32.
// A = ones(16x4) => D[m][n] += sum_k B[k][n]: column sums of 4 partial rows per
// WMMA, 3 WMMAs per iteration (sum_p / count / sum_t), accumulated in f32.
// B layout fact used: column index = lane%16; lanes<16 hold K rows {0,1},
// lanes>=16 hold K rows {2,3} (any K permutation is harmless under ones-A).
// D row 0 lives in accumulator element 0 of lanes 0..15.
// ---------------------------------------------------------------------------
__global__ __launch_bounds__(32) void ece_finalize_kernel(
    const float* __restrict__ partials, float* __restrict__ out,
    int nblocks, float total)
{
    const int lane = threadIdx.x;        // wave32, EXEC all ones (no divergence)
    const int col  = lane & 15;
    const int half = lane >> 4;

    v2f ones; ones[0] = 1.0f; ones[1] = 1.0f;
    v8f acc_p = {}; v8f acc_c = {}; v8f acc_t = {};

    for (int g = 0; g < nblocks; g += 4) {
        const float* r0 = partials + (size_t)(g + 2 * half + 0) * 48;
        const float* r1 = partials + (size_t)(g + 2 * half + 1) * 48;
        v2f bp; bp[0] = r0[col];      bp[1] = r1[col];
        v2f bc; bc[0] = r0[16 + col]; bc[1] = r1[16 + col];
        v2f bt; bt[0] = r0[32 + col]; bt[1] = r1[32 + col];
        acc_p = __builtin_amdgcn_wmma_f32_16x16x4_f32(false, ones, false, bp,
                                                      (short)0, acc_p, false, false);
        acc_c = __builtin_amdgcn_wmma_f32_16x16x4_f32(false, ones, false, bc,
                                                      (short)0, acc_c, false, false);
        acc_t = __builtin_amdgcn_wmma_f32_16x16x4_f32(false, ones, false, bt,
                                                      (short)0, acc_t, false, false);
    }

    __shared__ float contrib[16];
    if (lane < 16) {
        float sp  = acc_p[0];   // D[0][lane]
        float cnt = acc_c[0];
        float st  = acc_t[0];
        float e = 0.0f;
        if (lane < N_BINS && cnt > 0.0f) {
            float safe = cnt < 1.0f ? 1.0f : cnt;
            e = fabsf(sp / safe - st / safe) * (cnt / total);
        }
        contrib[lane] = e;
    }
    __syncthreads();
    if (lane == 0) {
        float s = 0.0f;
        for (int b = 0; b < N_BINS; ++b) s += contrib[b];
        out[0] = s;
    }
}

extern "C" void kernel_launch(void* const* d_in, const int* in_sizes, int n_in,
                              void* d_out, int out_size, void* d_ws, size_t ws_size,
                              hipStream_t stream) {
    const float* logits  = (const float*)d_in[0];
    const int*   targets = (const int*)d_in[1];
    const int n = in_sizes[0];

    float* partials = (float*)d_ws;   // GRID1 * 48 * 4 B = 384 KB scratch

    ece_partials_kernel<<<GRID1, BLOCK1, 0, stream>>>(logits, targets, partials, n);
    ece_finalize_kernel<<<1, 32, 0, stream>>>(partials, (float*)d_out, GRID1, (float)n);
}